// Gate_27066883899493
// MI455X (gfx1250) — compile-verified
//
#include <hip/hip_runtime.h>
#include <math.h>

// MoE gate: scores = sigmoid(x @ W^T) in fp32 via V_WMMA_F32_16X16X4_F32
// with double-buffered async global->LDS staging, then group-limited top-k.
//
// x:      [8192, 7168] fp32
// weight: [ 256, 7168] fp32
// bias:   [ 256]       fp32
// out:    weights [8192,8] fp32  ++  indices [8192,8] (stored as float)

typedef __attribute__((ext_vector_type(2))) float v2f;
typedef __attribute__((ext_vector_type(8))) float v8f;
typedef __attribute__((ext_vector_type(4))) int   v4i;

#define T_TOK 8192
#define DDIM  7168
#define EEXP  256
#define EPB   128         // experts per block (gridDim.y = 2)
#define KC    32          // K-chunk staged in LDS per pipeline stage
#define LDW   36          // padded LDS row stride (floats): rows 16B-aligned,
                          // 9*L mod 16 distinct -> conflict-free b64 frag reads
#define TOKS_PER_BLK 64
#define NGROUPS 8
#define GSIZE   32
#define TOPKG   4
#define TOPK    8
#define ROUTE_SCALE 2.5f

#if __has_builtin(__builtin_amdgcn_global_load_async_to_lds_b128)
#define HAVE_ASYNC_LDS 1
// Signature (from clang diagnostic): param0 = v4i addrspace(1)* (global src),
// param1 = v4i addrspace(3)* (LDS dst), then imm offset, imm cpol.
typedef __attribute__((address_space(1))) v4i* gv4i_ptr;
typedef __attribute__((address_space(3))) v4i* lv4i_ptr;
#endif

static __device__ __forceinline__ void async_stage_b128(const float* gsrc,
                                                        float* ldst) {
#ifdef HAVE_ASYNC_LDS
  // GLOBAL_LOAD_ASYNC_TO_LDS_B128: per-lane 16B global -> LDS, ASYNCcnt-tracked
  __builtin_amdgcn_global_load_async_to_lds_b128(
      (gv4i_ptr)(const_cast<float*>(gsrc)), (lv4i_ptr)(ldst), 0, 0);
#else
  *(float4*)ldst = *(const float4*)gsrc;
#endif
}

static __device__ __forceinline__ void wait_async_copies() {
#ifdef HAVE_ASYNC_LDS
#if __has_builtin(__builtin_amdgcn_s_wait_asynccnt)
  __builtin_amdgcn_s_wait_asynccnt(0);
#else
  asm volatile("s_wait_asynccnt 0" ::: "memory");
#endif
#endif
}

// ---------------------------------------------------------------------------
// Kernel 1: gate GEMM + sigmoid (fp32 WMMA, double-buffered LDS pipeline).
// Grid: (T/64, 256/EPB) blocks of 256 threads (8 waves).
// Wave (tt, eq): token tile tt = wave>>1 (16 tokens), expert quarter eq =
// wave&1 (64 experts -> 4 accumulator tiles of 16x16 f32).
// ---------------------------------------------------------------------------
__global__ __launch_bounds__(256) void gate_gemm_sigmoid(
    const float* __restrict__ x,
    const float* __restrict__ w,
    float* __restrict__ scores)
{
  __shared__ float xs[2][TOKS_PER_BLK * LDW];  // 2 x 64 x 36 floats (18 KB)
  __shared__ float wsh[2][EPB * LDW];          // 2 x 128 x 36 floats (36 KB)

  const int tid  = threadIdx.x;
  const int wave = tid >> 5;
  const int lane = tid & 31;
  const int half = lane >> 4;   // 0: K={0,1}, 1: K={2,3} (32-bit A/B layout)
  const int m    = lane & 15;   // M (A) / N (B) within the 16x16 tile

  const int tt = wave >> 1;     // token tile 0..3
  const int eq = wave & 1;      // expert half within block's 128 experts
  const long tokBase = (long)blockIdx.x * TOKS_PER_BLK;
  const int  expBase = blockIdx.y * EPB;

  v8f acc[4] = {};              // 4 x (16x16 f32) accumulators

  // Per-thread staging slots: weight 128x8 f4 -> 4/thread; x 64x8 f4 -> 2/thread
  int wrow[4], wc4[4];
#pragma unroll
  for (int it = 0; it < 4; ++it) {
    int idx = tid + it * 256;
    wrow[it] = idx >> 3;
    wc4[it]  = idx & 7;
  }
  int xrow[2], xc4[2];
#pragma unroll
  for (int it = 0; it < 2; ++it) {
    int idx = tid + it * 256;
    xrow[it] = idx >> 3;
    xc4[it]  = idx & 7;
  }

  // ---- prologue: stage chunk 0 into buffer 0 ----
#pragma unroll
  for (int it = 0; it < 4; ++it)
    async_stage_b128(&w[(long)(expBase + wrow[it]) * DDIM + wc4[it] * 4],
                     &wsh[0][wrow[it] * LDW + wc4[it] * 4]);
#pragma unroll
  for (int it = 0; it < 2; ++it)
    async_stage_b128(&x[(tokBase + xrow[it]) * DDIM + xc4[it] * 4],
                     &xs[0][xrow[it] * LDW + xc4[it] * 4]);
  wait_async_copies();
  __syncthreads();

  int p = 0;
  for (int k0 = 0; k0 < DDIM; k0 += KC) {
    const int np = p ^ 1;
    // ---- issue async prefetch of next chunk into the other buffer ----
    if (k0 + KC < DDIM) {
      const int kn = k0 + KC;
#pragma unroll
      for (int it = 0; it < 4; ++it)
        async_stage_b128(&w[(long)(expBase + wrow[it]) * DDIM + kn + wc4[it] * 4],
                         &wsh[np][wrow[it] * LDW + wc4[it] * 4]);
#pragma unroll
      for (int it = 0; it < 2; ++it)
        async_stage_b128(&x[(tokBase + xrow[it]) * DDIM + kn + xc4[it] * 4],
                         &xs[np][xrow[it] * LDW + xc4[it] * 4]);
    }

    // ---- compute: 8 K-steps x 4 expert tiles of v_wmma_f32_16x16x4_f32 ----
#pragma unroll
    for (int k = 0; k < KC; k += 4) {
      // A fragment (16x4): lane holds x[tok=m][k + 2*half + {0,1}]
      v2f a = *(const v2f*)&xs[p][(tt * 16 + m) * LDW + k + 2 * half];
#pragma unroll
      for (int et = 0; et < 4; ++et) {
        // B fragment (4x16): lane holds w[expert=N][k + 2*half + {0,1}]
        v2f b = *(const v2f*)
            &wsh[p][(eq * 64 + et * 16 + m) * LDW + k + 2 * half];
        acc[et] = __builtin_amdgcn_wmma_f32_16x16x4_f32(
            /*neg_a=*/false, a, /*neg_b=*/false, b,
            /*c_mod=*/(short)0, acc[et], /*reuse_a=*/false, /*reuse_b=*/false);
      }
    }

    // prefetched data must be LDS-visible to all waves before next chunk
    wait_async_copies();
    __syncthreads();
    p = np;
  }

  // ---- epilogue: sigmoid + store scores[T][E] ----
  // C/D layout: VGPR r -> (M = r + 8*half, N = m)
#pragma unroll
  for (int et = 0; et < 4; ++et) {
    int col = expBase + eq * 64 + et * 16 + m;
#pragma unroll
    for (int r = 0; r < 8; ++r) {
      long row = tokBase + tt * 16 + half * 8 + r;
      float v = acc[et][r];
      v = 1.0f / (1.0f + __expf(-v));
      scores[row * EEXP + col] = v;
    }
  }
}

// ---------------------------------------------------------------------------
// Kernel 2: group-limited routing.
// ---------------------------------------------------------------------------
__global__ __launch_bounds__(256) void route_topk(
    const float* __restrict__ scores,
    const float* __restrict__ bias,
    float* __restrict__ out_w,
    float* __restrict__ out_i)
{
  __shared__ float sc[32 * EEXP];   // 32 KB
  __shared__ float bsh[EEXP];

  const int tid  = threadIdx.x;
  const long tok0 = (long)blockIdx.x * 32;

  if (tid < EEXP) bsh[tid] = bias[tid];
#pragma unroll
  for (int it = 0; it < 32; ++it) {
    int idx = tid + it * 256;
    sc[idx] = scores[tok0 * EEXP + idx];
  }
  __syncthreads();

  if (tid < 32) {
    const float* s = &sc[tid * EEXP];
    const long t = tok0 + tid;

    // group score = sum of top-2 (score + bias) within each group of 32
    float gs[NGROUPS];
#pragma unroll
    for (int g = 0; g < NGROUPS; ++g) {
      float t1 = -INFINITY, t2 = -INFINITY;
      for (int e = 0; e < GSIZE; ++e) {
        float v = s[g * GSIZE + e] + bsh[g * GSIZE + e];
        if (v > t1) { t2 = t1; t1 = v; }
        else if (v > t2) { t2 = v; }
      }
      gs[g] = t1 + t2;
    }

    // top-4 groups (strict > keeps lowest index on ties, like jax top_k)
    unsigned keep = 0u;
#pragma unroll
    for (int j = 0; j < TOPKG; ++j) {
      int best = 0; float bv = -INFINITY;
      for (int g = 0; g < NGROUPS; ++g)
        if (!((keep >> g) & 1u) && gs[g] > bv) { bv = gs[g]; best = g; }
      keep |= 1u << best;
    }

    // top-8 experts among kept groups by (score + bias)
    float tv[TOPK]; int ti[TOPK];
#pragma unroll
    for (int i = 0; i < TOPK; ++i) { tv[i] = -INFINITY; ti[i] = 0; }
    for (int g = 0; g < NGROUPS; ++g) {
      if (!((keep >> g) & 1u)) continue;
      for (int e = 0; e < GSIZE; ++e) {
        int ei = g * GSIZE + e;
        float v = s[ei] + bsh[ei];
        if (v > tv[TOPK - 1]) {
          int pz = TOPK - 1;
          while (pz > 0 && v > tv[pz - 1]) {
            tv[pz] = tv[pz - 1]; ti[pz] = ti[pz - 1]; --pz;
          }
          tv[pz] = v; ti[pz] = ei;
        }
      }
    }

    // routing weights from ORIGINAL sigmoid scores (no bias), normalized
    float wv[TOPK]; float wsum = 0.f;
#pragma unroll
    for (int i = 0; i < TOPK; ++i) { wv[i] = s[ti[i]]; wsum += wv[i]; }
    const float inv = ROUTE_SCALE / wsum;
#pragma unroll
    for (int i = 0; i < TOPK; ++i) {
      out_w[t * TOPK + i] = wv[i] * inv;
      out_i[t * TOPK + i] = (float)ti[i];   // indices as output-dtype (f32)
    }
  }
}

// ---------------------------------------------------------------------------
extern "C" void kernel_launch(void* const* d_in, const int* in_sizes, int n_in,
                              void* d_out, int out_size, void* d_ws, size_t ws_size,
                              hipStream_t stream) {
  const float* x    = (const float*)d_in[0];
  const float* w    = (const float*)d_in[1];
  const float* bias = (const float*)d_in[2];
  float* out    = (float*)d_out;
  float* scores = (float*)d_ws;            // [8192, 256] fp32 = 8 MB scratch

  dim3 g1(T_TOK / TOKS_PER_BLK, EEXP / EPB);  // (128, 2) = 256 blocks
  gate_gemm_sigmoid<<<g1, 256, 0, stream>>>(x, w, scores);

  dim3 g2(T_TOK / 32);                        // 256 blocks
  route_topk<<<g2, 256, 0, stream>>>(scores, bias, out,
                                     out + (long)T_TOK * TOPK);
}